// GRU_42691974922588
// MI455X (gfx1250) — compile-verified
//
#include <hip/hip_runtime.h>
#include <math.h>

// Problem sizes (match reference).
constexpr int BSZ  = 1024;
constexpr int NIN  = 1024;
constexpr int NHID = 2048;
constexpr int NOUT = 32000;
constexpr int NG   = 2 * NHID;          // 4096 gate outputs
constexpr int KCAT = NIN + NHID;        // 3072

typedef float v2f __attribute__((ext_vector_type(2)));
typedef float v8f __attribute__((ext_vector_type(8)));

// ---------------------------------------------------------------------------
// Per-wave fp32 WMMA GEMM inner loop: accumulates an (MT*16)(M) x 64(N) tile
// of X(MxK, row-major) @ W(NxK, row-major)^T using V_WMMA_F32_16X16X4_F32.
//
// A fragment (16x4 f32): lane L holds row (mbase + (L&15)); VGPR0/1 hold the
// K-pair at offset (L>=16 ? 2 : 0) -> one contiguous 8-byte load per lane.
// B fragment (4x16 f32): lane L holds column n0+t*16+(L&15) of W^T, i.e. row
// n of W, same contiguous K-pair -> one 8-byte load per lane per tile.
// MT=2 -> 6 loads per 8 WMMAs (0.75 loads/WMMA) vs 1.25 at MT=1.
// ---------------------------------------------------------------------------
template <int MT>
__device__ __forceinline__ void wmma_gemm_tile(
    const float* __restrict__ X, int ldx,
    const float* __restrict__ W, int ldw,
    int K, int mrow0, int n0, int kin, int ln, v8f* acc)
{
    const float* xp[MT];
#pragma unroll
    for (int mt = 0; mt < MT; ++mt)
        xp[mt] = X + (size_t)(mrow0 + mt * 16) * ldx + kin;
    const float* wp[4];
#pragma unroll
    for (int t = 0; t < 4; ++t)
        wp[t] = W + (size_t)(n0 + t * 16 + ln) * ldw + kin;
#pragma unroll 4
    for (int k = 0; k < K; k += 4) {
        v2f a[MT], b[4];
#pragma unroll
        for (int mt = 0; mt < MT; ++mt) a[mt] = *(const v2f*)(xp[mt] + k);
#pragma unroll
        for (int t = 0; t < 4; ++t)     b[t]  = *(const v2f*)(wp[t] + k);
#pragma unroll
        for (int mt = 0; mt < MT; ++mt) {
#pragma unroll
            for (int t = 0; t < 4; ++t) {
                acc[mt * 4 + t] = __builtin_amdgcn_wmma_f32_16x16x4_f32(
                    false, a[mt], false, b[t], (short)0, acc[mt * 4 + t], false, false);
            }
        }
    }
}

// C/D layout (32-bit 16x16): VGPR v, lanes0-15 -> M=v, lanes16-31 -> M=v+8; N = lane&15.
// Workgroup = 8 waves arranged 2(M-groups) x 4(N-groups); block tile = MT*32 x 256.
#define WAVE_TILE_SETUP(MT)                                               \
    const int lane = threadIdx.x & 31;                                    \
    const int wave = threadIdx.x >> 5;                                    \
    const int ln   = lane & 15;                                           \
    const int kin  = (lane >> 4) * 2;                                     \
    const int hi8  = (lane >> 4) * 8;                                     \
    const int mbase = blockIdx.x * (32 * (MT)) + (wave >> 2) * (16 * (MT)); \
    const int n0    = blockIdx.y * 256 + (wave & 3) * 64;

// ---------------------------------------------------------------------------
// Kernel 1: gates = sigmoid([input, hidden] @ W_i2g^T + b_i2g)   (1024 x 4096)
// ---------------------------------------------------------------------------
__global__ __launch_bounds__(256)
void gru_gates_kernel(const float* __restrict__ input,
                      const float* __restrict__ hidden,
                      const float* __restrict__ W_i2g,
                      const float* __restrict__ b_i2g,
                      float* __restrict__ gates)
{
    WAVE_TILE_SETUP(2);
    v8f acc[8] = {};
    const int mrow0 = mbase + ln;
    // K split at the concat boundary: cols [0,1024) multiply input, [1024,3072) hidden.
    wmma_gemm_tile<2>(input,  NIN,  W_i2g,       KCAT, NIN,  mrow0, n0, kin, ln, acc);
    wmma_gemm_tile<2>(hidden, NHID, W_i2g + NIN, KCAT, NHID, mrow0, n0, kin, ln, acc);
#pragma unroll
    for (int mt = 0; mt < 2; ++mt) {
#pragma unroll
        for (int t = 0; t < 4; ++t) {
#pragma unroll
            for (int v = 0; v < 8; ++v) {
                const int m = mbase + mt * 16 + v + hi8;
                const int n = n0 + t * 16 + ln;
                const float x = acc[mt * 4 + t][v] + b_i2g[n];
                gates[(size_t)m * NG + n] = 1.0f / (1.0f + expf(-x));
            }
        }
    }
}

// ---------------------------------------------------------------------------
// Kernel 2: h = z*hidden + (1-z)*tanh(x@W_x2h^T + b + r*(h@W_h2h^T))  (1024 x 2048)
// Two accumulator sets -> keep MT=1 to stay within VGPR budget.
// ---------------------------------------------------------------------------
__global__ __launch_bounds__(256)
void gru_hcell_kernel(const float* __restrict__ input,
                      const float* __restrict__ hidden,
                      const float* __restrict__ W_x2h,
                      const float* __restrict__ b_x2h,
                      const float* __restrict__ W_h2h,
                      const float* __restrict__ gates,
                      float* __restrict__ h_out)
{
    WAVE_TILE_SETUP(1);
    v8f accx[4] = {};
    v8f acch[4] = {};
    const int mrow0 = mbase + ln;
    wmma_gemm_tile<1>(input,  NIN,  W_x2h, NIN,  NIN,  mrow0, n0, kin, ln, accx);
    wmma_gemm_tile<1>(hidden, NHID, W_h2h, NHID, NHID, mrow0, n0, kin, ln, acch);
#pragma unroll
    for (int t = 0; t < 4; ++t) {
#pragma unroll
        for (int v = 0; v < 8; ++v) {
            const int m = mbase + v + hi8;
            const int n = n0 + t * 16 + ln;
            const float z = gates[(size_t)m * NG + n];
            const float r = gates[(size_t)m * NG + NHID + n];
            const float hc = tanhf(accx[t][v] + b_x2h[n] + r * acch[t][v]);
            const float hp = hidden[(size_t)m * NHID + n];
            h_out[(size_t)m * NHID + n] = z * hp + (1.0f - z) * hc;
        }
    }
}

// ---------------------------------------------------------------------------
// Kernel 3: logits = h @ W_h2o^T + b_h2o   (1024 x 32000), into d_out front.
// blockIdx.x (M) is the fast launch dimension -> consecutive blocks share the
// same 2MB W_h2o tile in L2; W_h2o (262MB) streams from HBM exactly once.
// ---------------------------------------------------------------------------
__global__ __launch_bounds__(256)
void gru_logits_kernel(const float* __restrict__ h,
                       const float* __restrict__ W_h2o,
                       const float* __restrict__ b_h2o,
                       float* __restrict__ out)
{
    WAVE_TILE_SETUP(2);
    v8f acc[8] = {};
    const int mrow0 = mbase + ln;
    wmma_gemm_tile<2>(h, NHID, W_h2o, NHID, NHID, mrow0, n0, kin, ln, acc);
#pragma unroll
    for (int mt = 0; mt < 2; ++mt) {
#pragma unroll
        for (int t = 0; t < 4; ++t) {
#pragma unroll
            for (int v = 0; v < 8; ++v) {
                const int m = mbase + mt * 16 + v + hi8;
                const int n = n0 + t * 16 + ln;
                out[(size_t)m * NOUT + n] = acc[mt * 4 + t][v] + b_h2o[n];
            }
        }
    }
}

// ---------------------------------------------------------------------------
// Kernel 4: in-place log_softmax per row (wave32 shuffles + LDS reduction).
// ---------------------------------------------------------------------------
__global__ __launch_bounds__(256)
void gru_logsoftmax_kernel(float* __restrict__ out)
{
    __shared__ float sm[8];
    float* row = out + (size_t)blockIdx.x * NOUT;
    const int tid = threadIdx.x;

    float mx = -INFINITY;
    for (int i = tid; i < NOUT; i += 256) mx = fmaxf(mx, row[i]);
#pragma unroll
    for (int o = 16; o > 0; o >>= 1) mx = fmaxf(mx, __shfl_xor(mx, o, 32));
    if ((tid & 31) == 0) sm[tid >> 5] = mx;
    __syncthreads();
    mx = sm[0];
#pragma unroll
    for (int w = 1; w < 8; ++w) mx = fmaxf(mx, sm[w]);
    __syncthreads();

    float s = 0.0f;
    for (int i = tid; i < NOUT; i += 256) s += expf(row[i] - mx);
#pragma unroll
    for (int o = 16; o > 0; o >>= 1) s += __shfl_xor(s, o, 32);
    if ((tid & 31) == 0) sm[tid >> 5] = s;
    __syncthreads();
    s = 0.0f;
#pragma unroll
    for (int w = 0; w < 8; ++w) s += sm[w];

    const float lse = mx + logf(s);
    for (int i = tid; i < NOUT; i += 256) row[i] -= lse;
}

// ---------------------------------------------------------------------------
extern "C" void kernel_launch(void* const* d_in, const int* /*in_sizes*/, int /*n_in*/,
                              void* d_out, int /*out_size*/, void* d_ws, size_t /*ws_size*/,
                              hipStream_t stream)
{
    const float* input  = (const float*)d_in[0];
    const float* hidden = (const float*)d_in[1];
    const float* W_i2g  = (const float*)d_in[2];
    const float* b_i2g  = (const float*)d_in[3];
    const float* W_x2h  = (const float*)d_in[4];
    const float* b_x2h  = (const float*)d_in[5];
    const float* W_h2h  = (const float*)d_in[6];
    const float* W_h2o  = (const float*)d_in[7];
    const float* b_h2o  = (const float*)d_in[8];

    float* out   = (float*)d_out;                    // (1024, 32000) log-probs
    float* h_out = out + (size_t)BSZ * NOUT;         // (1024, 2048) new hidden
    float* gates = (float*)d_ws;                     // (1024, 4096) scratch, 16.8 MB

    const dim3 blk(256);
    gru_gates_kernel<<<dim3(BSZ / 64, NG / 256),    blk, 0, stream>>>(input, hidden, W_i2g, b_i2g, gates);
    gru_hcell_kernel<<<dim3(BSZ / 32, NHID / 256),  blk, 0, stream>>>(input, hidden, W_x2h, b_x2h, W_h2h, gates, h_out);
    gru_logits_kernel<<<dim3(BSZ / 64, NOUT / 256), blk, 0, stream>>>(h_out, W_h2o, b_h2o, out);
    gru_logsoftmax_kernel<<<dim3(BSZ), blk, 0, stream>>>(out);
}